// DTMLayer_73847667688044
// MI455X (gfx1250) — compile-verified
//
#include <hip/hip_runtime.h>
#include <hip/hip_bf16.h>
#include <stdint.h>
#include <math.h>

#define NPTS 4096   // H*W
#define NCH  32     // channels
#define NB   64     // coarse buckets: d2>>7, d2max=7938 -> 62
#define NFB  128    // fine bins within one coarse bucket
#define FSTR 129    // fine-hist padded stride (bank-conflict free)
#define CSTR 65     // cum padded stride (bank-conflict free)
#define SCRF 4224   // scratch floats: max(2*2112 for cumW/cumD, 32*129 fine hist)

typedef __attribute__((ext_vector_type(2))) float v2f;
typedef __attribute__((ext_vector_type(8))) float v8f;
typedef __attribute__((ext_vector_type(4))) unsigned int v4u;
typedef __attribute__((ext_vector_type(8))) int v8i;
typedef __attribute__((ext_vector_type(4))) int v4i;

// ---------------------------------------------------------------------------
// Kernel A: wb[c] = 0.05 * sum(weight[c, :])
// ---------------------------------------------------------------------------
__global__ __launch_bounds__(256) void dtm_wb_kernel(const float* __restrict__ weight,
                                                     float* __restrict__ wb) {
    __shared__ float red[256];
    const int c = blockIdx.x;
    float s = 0.f;
    for (int i = threadIdx.x; i < NPTS; i += 256) s += weight[c * NPTS + i];
    red[threadIdx.x] = s;
    __syncthreads();
    for (int off = 128; off > 0; off >>= 1) {
        if (threadIdx.x < off) red[threadIdx.x] += red[threadIdx.x + off];
        __syncthreads();
    }
    if (threadIdx.x == 0) wb[c] = 0.05f * red[0];
}

// ---------------------------------------------------------------------------
// Kernel B: one block per grid point j.
// ---------------------------------------------------------------------------
__global__ __launch_bounds__(256) void dtm_main_kernel(const float* __restrict__ weight,
                                                       const float* __restrict__ dist,
                                                       const float* __restrict__ wbws,
                                                       float* __restrict__ out) {
    __shared__ float sDist[NPTS];        // 16KB: dist row j (TDM target)
    __shared__ float sHW[NCH * NB];      // 8KB: coarse weight histogram
    __shared__ float sHD[NCH * NB];      // 8KB: coarse d2*weight histogram
    __shared__ float sScr[SCRF];         // 16.9KB: cumW/cumD, then fine hist
    __shared__ float sWb[NCH];
    __shared__ float sWbelow[NCH];
    __shared__ float sS1[NCH];
    __shared__ int   sBstar[NCH];

    const int tid  = threadIdx.x;
    const int wave = tid >> 5;
    const int lane = tid & 31;
    const int j    = blockIdx.x;

    float* cumW = sScr;          // 32 x CSTR
    float* cumD = sScr + 2112;   // 32 x CSTR
    float* fhw  = sScr;          // 32 x FSTR (aliases cum*, used later)

    // ---- Stage dist row j (16KB) into LDS via Tensor Data Mover -----------
#if __has_builtin(__builtin_amdgcn_tensor_load_to_lds)
    if (wave == 0) {
        uint64_t ga = (uint64_t)(uintptr_t)(dist + (size_t)j * NPTS);
        uint32_t la = (uint32_t)(uintptr_t)(&sDist[0]);
        v4u g0; v8i g1; v4i g2; v4i g3; v8i g4;
        g0[0] = 1u;                                                  // count=1
        g0[1] = la;                                                  // lds_addr
        g0[2] = (uint32_t)ga;                                        // global_addr lo
        g0[3] = (uint32_t)((ga >> 32) & 0x1FFFFFFull) | (2u << 30);  // addr[56:32] | type=2
        g1[0] = (int)(2u << 16);                    // data_size=2 (4 bytes/elem)
        g1[1] = (int)((NPTS & 0xFFFF) << 16);       // tensor_dim0 = 4096
        g1[2] = (int)(1u << 16);                    // tensor_dim1 = 1
        g1[3] = (int)((NPTS & 0xFFFF) << 16);       // tile_dim0 = 4096
        g1[4] = 1;                                  // tile_dim1 = 1
        g1[5] = NPTS;                               // tensor_dim0_stride
        g1[6] = 0; g1[7] = 0;
        g2[0] = g2[1] = g2[2] = g2[3] = 0;
        g3[0] = g3[1] = g3[2] = g3[3] = 0;
        g4[0] = g4[1] = g4[2] = g4[3] = 0;
        g4[4] = g4[5] = g4[6] = g4[7] = 0;
        __builtin_amdgcn_tensor_load_to_lds(g0, g1, g2, g3, g4, 0);
    }
#else
    for (int i = tid; i < NPTS; i += 256) sDist[i] = dist[(size_t)j * NPTS + i];
#endif

    // zero coarse hists / load budgets while the DMA is in flight
    for (int i = tid; i < NCH * NB; i += 256) { sHW[i] = 0.f; sHD[i] = 0.f; }
    if (tid < NCH) sWb[tid] = wbws[tid];

#if __has_builtin(__builtin_amdgcn_tensor_load_to_lds)
    if (wave == 0) __builtin_amdgcn_s_wait_tensorcnt(0);
#endif
    __syncthreads();

    // ---- Pass 1: coarse weighted histograms (lanes = neighbors) -----------
    for (int ib = wave; ib < NPTS / 32; ib += 8) {
        const int i = ib * 32 + lane;
        const float d = sDist[i];
        const int d2  = (int)fmaf(d, d, 0.5f);   // exact integer lattice d^2
        const int cb  = d2 >> 7;
        const float d2f = (float)d2;
        for (int c = 0; c < NCH; ++c) {
            const float w = weight[c * NPTS + i];        // coalesced
            atomicAdd(&sHW[c * NB + cb], w);             // ds_add_f32
            atomicAdd(&sHD[c * NB + cb], d2f * w);       // ds_add_f32
        }
    }
    __syncthreads();

    // ---- Cumulative sums over buckets: H(32x64) x upper-triangular ones ---
#if __has_builtin(__builtin_amdgcn_wmma_f32_16x16x4_f32)
    if (wave < 4) {
        const int nt    = wave;
        const int mrow  = lane & 15;
        const int khalf = (lane >> 4) << 1;   // lanes 16-31 hold K+2,K+3
        const int hi8   = (lane >> 4) << 3;   // D rows +8 for upper half
        const int n     = nt * 16 + mrow;     // global column (bucket)
        for (int stat = 0; stat < 2; ++stat) {
            const float* H = stat ? sHD : sHW;
            float*       C = stat ? cumD : cumW;
            for (int mt = 0; mt < 2; ++mt) {
                v8f acc = {0.f, 0.f, 0.f, 0.f, 0.f, 0.f, 0.f, 0.f};
                const int m = mt * 16 + mrow;
                for (int kt = 0; kt < 16; ++kt) {
                    const int k0 = 4 * kt + khalf;
                    v2f av = { H[m * NB + k0], H[m * NB + k0 + 1] };
                    v2f bv = { (k0     <= n) ? 1.0f : 0.0f,
                               (k0 + 1 <= n) ? 1.0f : 0.0f };
                    acc = __builtin_amdgcn_wmma_f32_16x16x4_f32(
                        false, av, false, bv, (short)0, acc, false, false);
                }
                for (int v = 0; v < 8; ++v)
                    C[(mt * 16 + v + hi8) * CSTR + n] = acc[v];
            }
        }
    }
#else
    if (wave == 0) {
        float aW = 0.f, aD = 0.f;
        for (int b = 0; b < NB; ++b) {
            aW += sHW[lane * NB + b];
            aD += sHD[lane * NB + b];
            cumW[lane * CSTR + b] = aW;
            cumD[lane * CSTR + b] = aD;
        }
    }
#endif
    __syncthreads();

    // ---- Find crossing coarse bucket per channel (lane = channel) ---------
    if (wave == 0) {
        const int c = lane;
        const float wbv = sWb[c];
        int bs = NB - 2;                      // fallback: last real bucket (62)
        float below = 0.f, s1 = 0.f;
        float prevW = 0.f, prevD = 0.f;
        bool found = false;
        for (int b = 0; b < 63; ++b) {
            const float cw = cumW[c * CSTR + b];
            const float cd = cumD[c * CSTR + b];
            if (!found && cw >= wbv) {
                bs = b; below = prevW; s1 = prevD; found = true;
            }
            prevW = cw; prevD = cd;
        }
        if (!found) { below = prevW; s1 = prevD; }
        sBstar[c]  = bs;
        sWbelow[c] = below;
        sS1[c]     = s1;
    }
    __syncthreads();

    // ---- zero fine histogram (aliases cum arrays, no longer needed) -------
    for (int i = tid; i < NCH * FSTR; i += 256) fhw[i] = 0.f;
    __syncthreads();

    // ---- Pass 2: fine histogram inside each channel's crossing bucket -----
    {
        const int myBs = sBstar[lane];
        for (int ib = wave; ib < NPTS / 32; ib += 8) {
            const int i = ib * 32 + lane;
            const float d = sDist[i];
            const int d2  = (int)fmaf(d, d, 0.5f);
            const int cb  = d2 >> 7;
            const int fb  = d2 & 127;
            for (int c = 0; c < NCH; ++c) {
                const int bsc = __shfl(myBs, c, 32);
                if (cb == bsc)
                    atomicAdd(&fhw[c * FSTR + fb], weight[c * NPTS + i]);
            }
        }
    }
    __syncthreads();

    // ---- Final scan of 128 fine bins; emit result (lane = channel) --------
    if (wave == 0) {
        const int c = lane;
        const float wbv = sWb[c];
        float accW = sWbelow[c];
        float s1   = sS1[c];
        const int bs = sBstar[c];
        float dtm = -1.f;
        for (int fb = 0; fb < NFB; ++fb) {
            const float fw  = fhw[c * FSTR + fb];
            const float d2v = (float)(bs * 128 + fb);
            const float nw  = accW + fw;
            if (dtm < 0.f && nw >= wbv)
                dtm = s1 + d2v * (wbv - accW);   // fractional last bucket
            s1   = fmaf(d2v, fw, s1);
            accW = nw;
        }
        if (dtm < 0.f) dtm = s1;                 // numeric-slop fallback
        const float r = (wbv > 0.f) ? sqrtf(dtm / wbv) : 0.f;
        out[c * NPTS + j] = r;
    }
}

// ---------------------------------------------------------------------------
extern "C" void kernel_launch(void* const* d_in, const int* in_sizes, int n_in,
                              void* d_out, int out_size, void* d_ws, size_t ws_size,
                              hipStream_t stream) {
    const float* weight = (const float*)d_in[0];   // (32, 4096) f32
    const float* dist   = (const float*)d_in[1];   // (4096, 4096) f32
    float* wb  = (float*)d_ws;                     // 32 floats scratch
    float* out = (float*)d_out;                    // (32, 4096) f32

    dtm_wb_kernel<<<NCH, 256, 0, stream>>>(weight, wb);
    dtm_main_kernel<<<NPTS, 256, 0, stream>>>(weight, dist, wb, out);
}